// GraphLossWPhysicsLoss_60644938219636
// MI455X (gfx1250) — compile-verified
//
#include <hip/hip_runtime.h>
#include <math.h>

// ---------------------------------------------------------------------------
// GraphLossWPhysicsLoss for MI455X (gfx1250, wave32).
// Edge scatter = L2-resident f32 atomics; spectral 2D DFT = f32 WMMA GEMMs.
// ---------------------------------------------------------------------------

typedef __attribute__((ext_vector_type(2))) float v2f;
typedef __attribute__((ext_vector_type(8))) float v8f;

#define EPSF 1e-8f
constexpr int Bc = 4;        // batches
constexpr int Nc = 100000;   // nodes per batch
constexpr int Ec = 1600000;  // edges per batch
constexpr int Gc = 64;       // raster grid

__device__ __forceinline__ void atomAdd(float* p, float v) { unsafeAtomicAdd(p, v); }

// D = A(16x4) * B(4x16) + C, f32, wave32
__device__ __forceinline__ v8f wmma4(v2f a, v2f b, v8f c) {
  return __builtin_amdgcn_wmma_f32_16x16x4_f32(false, a, false, b, (short)0, c,
                                               false, false);
}

// twiddle: theta = 2*pi*((k*n) mod 64)/64 ; returns c=cos, s=sin
__device__ __forceinline__ void wangle(int k, int n, float& c, float& s) {
  int m = (k * n) & 63;
  float t = (float)m * 0.09817477042468103f; // 2*pi/64
  __sincosf(t, &s, &c);
}

// ---------------------------------------------------------------------------
__global__ void k_zero(float* wsf, int nz, int* mm, float* out) {
  int i = blockIdx.x * blockDim.x + threadIdx.x;
  if (i < nz) wsf[i] = 0.f;
  if (i < 8) {               // per-batch, per-component min/max (bit-atomics)
    mm[i] = 0x7f7fffff;      // +FLT_MAX bits (x >= 0)
    mm[8 + i] = 0;           // 0.0f bits
  }
  if (i == 0) out[0] = 0.f;
}

__global__ void k_minmax(const float* __restrict__ x, int* mm, int BN) {
  int g = blockIdx.x * blockDim.x + threadIdx.x;
  if (g >= BN) return;
  int b = g / Nc;
  int b0 = __float_as_int(x[2 * g + 0]);
  int b1 = __float_as_int(x[2 * g + 1]);
  atomicMin(&mm[b * 2 + 0], b0);
  atomicMin(&mm[b * 2 + 1], b1);
  atomicMax(&mm[8 + b * 2 + 0], b0);
  atomicMax(&mm[8 + b * 2 + 1], b1);
}

// Scatter per-edge contributions into 8 per-node accumulators:
// [0..3]=grad num (Jxx,Jxy,Jyx,Jyy), [4]=sum w, [5..6]=lap num, [7]=sum w2
__global__ void k_edges(const float* __restrict__ x,
                        const float* __restrict__ pred,
                        const int* __restrict__ edg, float* acc, int BE) {
  int ge = blockIdx.x * blockDim.x + threadIdx.x;
  if (ge >= BE) return;
  int b = ge / Ec;
  int i = edg[2 * ge + 0];
  int j = edg[2 * ge + 1];
  int gi = b * Nc + i, gj = b * Nc + j;
  float xi0 = x[2 * gi], xi1 = x[2 * gi + 1];
  float xj0 = x[2 * gj], xj1 = x[2 * gj + 1];
  float dx = xj0 - xi0, dy = xj1 - xi1;          // recv - send
  float len2 = dx * dx + dy * dy + EPSF;
  float inv = rsqrtf(len2);
  float ux = dx * inv, uy = dy * inv;
  float w = 1.0f / len2;                          // grad weight
  float w2 = 1.0f / (len2 + EPSF);                // laplacian weight (2x eps)
  float pi0 = pred[2 * gi], pi1 = pred[2 * gi + 1];
  float pj0 = pred[2 * gj], pj1 = pred[2 * gj + 1];
  float d0 = pj0 - pi0, d1 = pj1 - pi1;
  float c00 = w * d0 * ux, c01 = w * d0 * uy;     // grad u_x
  float c10 = w * d1 * ux, c11 = w * d1 * uy;     // grad u_y
  float l0 = d0 * w2, l1 = d1 * w2;               // laplacian
  float* ai = acc + (size_t)gi * 8;
  float* aj = acc + (size_t)gj * 8;
  atomAdd(ai + 0, c00);  atomAdd(ai + 1, c01);
  atomAdd(ai + 2, c10);  atomAdd(ai + 3, c11);
  atomAdd(ai + 4, w);
  atomAdd(ai + 5, l0);   atomAdd(ai + 6, l1);
  atomAdd(ai + 7, w2);
  atomAdd(aj + 0, -c00); atomAdd(aj + 1, -c01);
  atomAdd(aj + 2, -c10); atomAdd(aj + 3, -c11);
  atomAdd(aj + 4, w);
  atomAdd(aj + 5, -l0);  atomAdd(aj + 6, -l1);
  atomAdd(aj + 7, w2);
}

// Per-node finalize: MSE + div^2 + momentum residual^2, block-reduced into out
__global__ void k_node(const float* __restrict__ pred,
                       const float* __restrict__ target,
                       const float* __restrict__ acc, float* out, int BN) {
  __shared__ float red[256];
  int g = blockIdx.x * blockDim.x + threadIdx.x;
  float val = 0.f;
  if (g < BN) {
    const float* a = acc + (size_t)g * 8;
    float den = a[4] + EPSF;
    float Jxx = a[0] / den, Jxy = a[1] / den;
    float Jyx = a[2] / den, Jyy = a[3] / den;
    float dl = a[7] + EPSF;
    float lapx = a[5] / dl, lapy = a[6] / dl;
    float px = pred[2 * g], py = pred[2 * g + 1];
    float tx = target[2 * g], ty = target[2 * g + 1];
    float dv = Jxx + Jyy;
    float advx = Jxx * px + Jxy * py;
    float advy = Jyx * px + Jyy * py;
    float rx = (px - tx) + advx - lapx;   // DT = 1
    float ry = (py - ty) + advy - lapy;
    float ex = px - tx, ey = py - ty;
    val = (1.0f / 800000.f) * (ex * ex + ey * ey)   // MSE_W / (B*N*2)
        + (0.1f / 400000.f) * (dv * dv)             // DIV_W / (B*N)
        + (0.1f / 800000.f) * (rx * rx + ry * ry);  // MOM_W / (B*N*2)
  }
  red[threadIdx.x] = val;
  __syncthreads();
  for (int s = 128; s > 0; s >>= 1) {
    if (threadIdx.x < s) red[threadIdx.x] += red[threadIdx.x + s];
    __syncthreads();
  }
  if (threadIdx.x == 0) atomAdd(out, red[0]);
}

__global__ void k_raster(const float* __restrict__ x,
                         const float* __restrict__ pred,
                         const int* __restrict__ mm, float* Vg, float* Cg,
                         int BN) {
  int g = blockIdx.x * blockDim.x + threadIdx.x;
  if (g >= BN) return;
  int b = g / Nc;
  float xmin0 = __int_as_float(mm[b * 2 + 0]);
  float xmin1 = __int_as_float(mm[b * 2 + 1]);
  float xmax0 = __int_as_float(mm[8 + b * 2 + 0]);
  float xmax1 = __int_as_float(mm[8 + b * 2 + 1]);
  float r0 = fmaxf(xmax0 - xmin0, 1e-6f);
  float r1 = fmaxf(xmax1 - xmin1, 1e-6f);
  float xn0 = (x[2 * g + 0] - xmin0) / r0;
  float xn1 = (x[2 * g + 1] - xmin1) / r1;
  int gx = (int)(xn0 * 63.f); gx = gx < 0 ? 0 : (gx > 63 ? 63 : gx);
  int gy = (int)(xn1 * 63.f); gy = gy < 0 ? 0 : (gy > 63 ? 63 : gy);
  int cell = gy * Gc + gx;
  atomAdd(&Vg[((size_t)b * 4096 + cell) * 2 + 0], pred[2 * g + 0]);
  atomAdd(&Vg[((size_t)b * 4096 + cell) * 2 + 1], pred[2 * g + 1]);
  atomAdd(&Cg[(size_t)b * 4096 + cell], 1.f);
}

// One block per batch. 2D rDFT via f32 WMMA: F = W * (V * W), W symmetric.
__global__ void k_spectrum(const float* __restrict__ Vg,
                           const float* __restrict__ Cg, float* out) {
  __shared__ float Vl[64][65];
  __shared__ float Tre[64][65];
  __shared__ float Tim[64][65];
  __shared__ float Pp[64][34];
  __shared__ float rnum[31], rden[31], ek[31], cb[31];
  const int tid = threadIdx.x;
  const int b = blockIdx.x;
  for (int i = tid; i < 64 * 34; i += 512) Pp[i / 34][i % 34] = 0.f;
  if (tid < 31) { rnum[tid] = 0.f; rden[tid] = 0.f; }
  __syncthreads();

  const int w = tid >> 5, lane = tid & 31;
  const int ti = w >> 2, tj = w & 3;          // 16 waves -> 4x4 output tiles
  const int rowA = ti * 16 + (lane & 15);     // A-frag row (M)
  const int colB = tj * 16 + (lane & 15);     // B-frag col (N)
  const int khalf = (lane >> 4) << 1;         // K sub-offset per half-wave
  const int rbD = ti * 16 + ((lane >> 4) << 3); // D-frag row base
  const int colD = tj * 16 + (lane & 15);

  for (int ch = 0; ch < 2; ++ch) {
    for (int idx = tid; idx < 4096; idx += 512)
      Vl[idx >> 6][idx & 63] =
          Vg[((size_t)b * 4096 + idx) * 2 + ch] /
          (Cg[(size_t)b * 4096 + idx] + EPSF);
    __syncthreads();

    // stage 1: T = V * W  (T_re = V*cos, T_im = -V*sin)
    v8f aR = {}, aI = {};
    for (int kk = 0; kk < 16; ++kk) {
      int kb = kk * 4 + khalf;
      v2f a;  a.x = Vl[rowA][kb]; a.y = Vl[rowA][kb + 1];
      float c0, s0, c1, s1;
      wangle(kb, colB, c0, s0);
      wangle(kb + 1, colB, c1, s1);
      v2f bR; bR.x = c0;  bR.y = c1;
      v2f bI; bI.x = -s0; bI.y = -s1;
      aR = wmma4(a, bR, aR);
      aI = wmma4(a, bI, aI);
    }
    for (int r = 0; r < 8; ++r) {
      Tre[rbD + r][colD] = aR[r];
      Tim[rbD + r][colD] = aI[r];
    }
    __syncthreads();

    // stage 2: F = W * T ; F_re = cos*Tre + sin*Tim ; F_im = cos*Tim - sin*Tre
    v8f fR = {}, fI = {};
    for (int kk = 0; kk < 16; ++kk) {
      int kb = kk * 4 + khalf;
      float c0, s0, c1, s1;
      wangle(rowA, kb, c0, s0);
      wangle(rowA, kb + 1, c1, s1);
      v2f aC; aC.x = c0;  aC.y = c1;
      v2f aP; aP.x = s0;  aP.y = s1;
      v2f aM; aM.x = -s0; aM.y = -s1;
      v2f bRe; bRe.x = Tre[kb][colB]; bRe.y = Tre[kb + 1][colB];
      v2f bIm; bIm.x = Tim[kb][colB]; bIm.y = Tim[kb + 1][colB];
      fR = wmma4(aC, bRe, fR);
      fR = wmma4(aP, bIm, fR);
      fI = wmma4(aC, bIm, fI);
      fI = wmma4(aM, bRe, fI);
    }
    for (int r = 0; r < 8; ++r) {
      int ky = rbD + r, kx = colD;
      if (kx < 33) Pp[ky][kx] += 0.5f * (fR[r] * fR[r] + fI[r] * fI[r]);
    }
    __syncthreads();
  }

  // radial binning over [64 x 33] power grid, 31 bins
  const float step = (0.70710678f + 1e-6f) / 31.f;
  for (int idx = tid; idx < 64 * 33; idx += 512) {
    int iy = idx / 33, ix = idx % 33;
    float fy = (float)(iy < 32 ? iy : iy - 64) * (1.f / 64.f);
    float fx = (float)ix * (1.f / 64.f);
    float R = sqrtf(fx * fx + fy * fy);
    int id = (int)(R / step);
    id = id < 0 ? 0 : (id > 30 ? 30 : id);
    atomicAdd(&rnum[id], Pp[iy][ix]);
    atomicAdd(&rden[id], 1.f);
  }
  __syncthreads();
  if (tid < 31) ek[tid] = fmaxf(rnum[tid] / (rden[tid] + EPSF), 1e-12f);
  __syncthreads();
  if (tid < 31) {
    float el = tid > 0 ? ek[tid - 1] : 0.f;
    float er = tid < 30 ? ek[tid + 1] : 0.f;
    float es = (el + ek[tid] + er) * (1.f / 3.f);
    float d = logf(ek[tid]) - logf(es);
    cb[tid] = d * d;
  }
  __syncthreads();
  if (tid == 0) {
    float s = 0.f;
    for (int i = 0; i < 31; ++i) s += cb[i];
    atomAdd(out, s * (0.01f / 124.f));   // SPEC_W / (B*31)
  }
}

// ---------------------------------------------------------------------------
extern "C" void kernel_launch(void* const* d_in, const int* in_sizes, int n_in,
                              void* d_out, int out_size, void* d_ws,
                              size_t ws_size, hipStream_t stream) {
  (void)in_sizes; (void)n_in; (void)out_size; (void)ws_size;
  const float* pred   = (const float*)d_in[0];  // [B,N,2]
  const float* target = (const float*)d_in[1];  // [B,N,2]
  const float* x      = (const float*)d_in[2];  // [B,N,2]
  const int*   edges  = (const int*)d_in[3];    // [B,E,2]
  float* out = (float*)d_out;
  float* wsf = (float*)d_ws;

  const int accN = Bc * Nc * 8;        // 3,200,000 floats (12.8 MB)
  const int vgN  = Bc * Gc * Gc * 2;   // 32,768
  const int cgN  = Bc * Gc * Gc;       // 16,384
  const int nz   = accN + vgN + cgN;
  float* acc = wsf;
  float* Vg  = wsf + accN;
  float* Cg  = Vg + vgN;
  int*   mm  = (int*)(Cg + cgN);       // 16 ints (min/max bits)

  const int BN = Bc * Nc, BE = Bc * Ec;

  k_zero<<<(nz + 255) / 256, 256, 0, stream>>>(wsf, nz, mm, out);
  k_minmax<<<(BN + 255) / 256, 256, 0, stream>>>(x, mm, BN);
  k_edges<<<(BE + 255) / 256, 256, 0, stream>>>(x, pred, edges, acc, BE);
  k_node<<<(BN + 255) / 256, 256, 0, stream>>>(pred, target, acc, out, BN);
  k_raster<<<(BN + 255) / 256, 256, 0, stream>>>(x, pred, mm, Vg, Cg, BN);
  k_spectrum<<<Bc, 512, 0, stream>>>(Vg, Cg, out);
}